// DFA_3770981285988
// MI455X (gfx1250) — compile-verified
//
#include <hip/hip_runtime.h>
#include <stdint.h>

// Soft-DFA scan: q_{t+1} = delta[sym_t] @ q_t, 4096 steps, n=512, fp32.
// delta (128 MB) is L2-resident on MI455X (192 MB L2). Persistent 32-WG grid,
// one global barrier per step, rows partitioned 16/WG. delta[sym_{t+1}] tile is
// prefetched into LDS with CDNA5 async-to-LDS loads while step t computes with
// fp32 WMMA (v_wmma_f32_16x16x4_f32, B = q broadcast across all 16 columns).

#define N_STATES   512
#define SEQ_LEN    4096
#define NWG        32
#define ROWS_PER_WG 16            // 512 / 32
#define TPB        256            // 8 wave32s
#define WAVES      8
#define PAD_STRIDE 516            // padded LDS row (floats): conflict-free A loads
#define CHUNKS_PER_WAVE 16        // 128 K-chunks of 4 / 8 waves

typedef float v2f __attribute__((ext_vector_type(2)));
typedef float v8f __attribute__((ext_vector_type(8)));

// Copy this WG's contiguous 32KB slice of delta[sym] into (row-padded) LDS.
// 8 async b128 instructions per wave (32 lanes x 16B each = 512B/instr).
__device__ __forceinline__ void async_copy_tile(const float* gbase,
                                                uint32_t lds_base, int tid) {
#pragma unroll
  for (int i = 0; i < 8; ++i) {
    int c   = i * TPB + tid;      // 0..2047 : 16B chunk index
    int row = c >> 7;             // 0..15
    int col = c & 127;            // 0..127 (16B units)
    uint64_t g = (uint64_t)(uintptr_t)gbase + (uint64_t)row * 2048u + (uint64_t)col * 16u;
    uint32_t l = lds_base + (uint32_t)row * (PAD_STRIDE * 4u) + (uint32_t)col * 16u;
    asm volatile("global_load_async_to_lds_b128 %0, %1, off"
                 :: "v"(l), "v"(g) : "memory");
  }
}

__global__ void __launch_bounds__(TPB)
dfa_init_kernel(float* q_bufs, int* cnt) {
  int tid = threadIdx.x + blockIdx.x * blockDim.x;
  if (tid == 0) *cnt = 0;                      // reset grid barrier each launch
  if (tid < N_STATES) q_bufs[tid] = (tid == 0) ? 1.0f : 0.0f;  // q0 = e_0 (parity 0)
}

__global__ void __launch_bounds__(TPB)
dfa_scan_kernel(const int* __restrict__ syms, const float* __restrict__ delta,
                float* q_bufs, int* barrier_cnt) {
  __shared__ float tile[2][ROWS_PER_WG * PAD_STRIDE];  // double-buffered matrix slice
  __shared__ float q_lds[N_STATES];
  __shared__ float partial[WAVES][ROWS_PER_WG];

  const int tid  = threadIdx.x;
  const int wg   = blockIdx.x;
  const int wave = tid >> 5;
  const int lane = tid & 31;
  const int half = lane >> 4;   // A fragment: lanes 0-15 hold K={0,1}, 16-31 hold K={2,3}
  const int lrow = lane & 15;   // A row M within the 16-row tile

  const uint32_t tile_lds0 = (uint32_t)(uintptr_t)&tile[0][0];
  const uint32_t tile_lds1 = (uint32_t)(uintptr_t)&tile[1][0];
  const size_t   row_off   = (size_t)wg * ROWS_PER_WG * N_STATES;

  // Prologue: prefetch tile for step 0 into buffer 0.
  async_copy_tile(delta + (size_t)syms[0] * N_STATES * N_STATES + row_off,
                  tile_lds0, tid);

  for (int t = 0; t < SEQ_LEN; ++t) {
    __syncthreads();                       // all waves done with previous step
    if (t > 0) {                           // grid barrier: q_t fully published
      if (tid == 0) {
        __hip_atomic_fetch_add(barrier_cnt, 1, __ATOMIC_RELEASE,
                               __HIP_MEMORY_SCOPE_AGENT);
        const int target = t * NWG;
        while (__hip_atomic_load(barrier_cnt, __ATOMIC_ACQUIRE,
                                 __HIP_MEMORY_SCOPE_AGENT) < target)
          __builtin_amdgcn_s_sleep(2);
      }
      __syncthreads();
    }

    // q_t (global, written by remote WGPs) -> LDS via agent-scoped loads (L2-fresh).
    {
      const float* qcur = q_bufs + (size_t)(t & 1) * N_STATES;
      int v0 = __hip_atomic_load((const int*)(qcur + tid),       __ATOMIC_RELAXED,
                                 __HIP_MEMORY_SCOPE_AGENT);
      int v1 = __hip_atomic_load((const int*)(qcur + tid + TPB), __ATOMIC_RELAXED,
                                 __HIP_MEMORY_SCOPE_AGENT);
      q_lds[tid]       = __int_as_float(v0);
      q_lds[tid + TPB] = __int_as_float(v1);
    }

    // Prefetch delta[sym_{t+1}] into the other LDS buffer, then wait for tile t:
    // each wave has <=16 async instrs in flight; waiting to <=8 retires tile t.
    if (t + 1 < SEQ_LEN) {
      async_copy_tile(delta + (size_t)syms[t + 1] * N_STATES * N_STATES + row_off,
                      ((t + 1) & 1) ? tile_lds1 : tile_lds0, tid);
      asm volatile("s_wait_asynccnt 8" ::: "memory");
    } else {
      asm volatile("s_wait_asynccnt 0" ::: "memory");
    }
    __syncthreads();                       // tile t complete for every wave

    // fp32 WMMA matvec: D(16x16) += A(16x4 of delta) x B(4x16 = q broadcast).
    // Every output column computes the same 16 partial sums; chunks of K split
    // across waves, chained through the accumulator.
    const float* tl = &tile[t & 1][0];
    v8f acc = {0.f, 0.f, 0.f, 0.f, 0.f, 0.f, 0.f, 0.f};
#pragma unroll
    for (int cc = 0; cc < CHUNKS_PER_WAVE; ++cc) {
      const int j0 = (wave * CHUNKS_PER_WAVE + cc) * 4;
      const int kk = j0 + 2 * half;
      v2f a, b;
      a.x = tl[lrow * PAD_STRIDE + kk];     // conflict-free: padded stride
      a.y = tl[lrow * PAD_STRIDE + kk + 1];
      b.x = q_lds[kk];                      // broadcast within each half-wave
      b.y = q_lds[kk + 1];
      acc = __builtin_amdgcn_wmma_f32_16x16x4_f32(
          /*neg_a=*/false, a, /*neg_b=*/false, b,
          /*c_mod=*/(short)0, acc, /*reuse_a=*/false, /*reuse_b=*/false);
    }
    // C/D layout: VGPR r -> M=r (lanes 0-15), M=8+r (lanes 16-31); all N equal.
    if (lrow == 0) {
#pragma unroll
      for (int r = 0; r < 8; ++r) partial[wave][half * 8 + r] = acc[r];
    }
    __syncthreads();

    // Cross-wave reduce and publish q_{t+1} slice (agent scope, fenced).
    if (tid < ROWS_PER_WG) {
      float s = 0.f;
#pragma unroll
      for (int w = 0; w < WAVES; ++w) s += partial[w][tid];
      float* qn = q_bufs + (size_t)((t + 1) & 1) * N_STATES + wg * ROWS_PER_WG + tid;
      __hip_atomic_store((int*)qn, __float_as_int(s), __ATOMIC_RELEASE,
                         __HIP_MEMORY_SCOPE_AGENT);
      __threadfence();                     // globally visible before next arrival
    }
  }
}

__global__ void __launch_bounds__(TPB)
dfa_dot_kernel(const float* __restrict__ q, const float* __restrict__ f,
               float* __restrict__ out) {
  __shared__ float red[TPB];
  int tid = threadIdx.x;
  red[tid] = q[tid] * f[tid] + q[tid + TPB] * f[tid + TPB];
  __syncthreads();
  for (int off = TPB / 2; off > 0; off >>= 1) {
    if (tid < off) red[tid] += red[tid + off];
    __syncthreads();
  }
  if (tid == 0) out[0] = red[0];
}

extern "C" void kernel_launch(void* const* d_in, const int* in_sizes, int n_in,
                              void* d_out, int out_size, void* d_ws, size_t ws_size,
                              hipStream_t stream) {
  const int*   syms  = (const int*)d_in[0];
  const float* delta = (const float*)d_in[1];
  const float* f     = (const float*)d_in[2];
  float*       out   = (float*)d_out;

  int*   cnt    = (int*)d_ws;                       // grid-barrier counter
  float* q_bufs = (float*)((char*)d_ws + 128);      // 2 x 512 floats (ping-pong)

  dfa_init_kernel<<<1, TPB, 0, stream>>>(q_bufs, cnt);
  dfa_scan_kernel<<<NWG, TPB, 0, stream>>>(syms, delta, q_bufs, cnt);
  // SEQ_LEN is even -> q_final lives in parity-0 buffer.
  dfa_dot_kernel<<<1, TPB, 0, stream>>>(q_bufs, f, out);
}